// Model_446676599185
// MI455X (gfx1250) — compile-verified
//
#include <hip/hip_runtime.h>

typedef __attribute__((ext_vector_type(2))) float v2f;
typedef __attribute__((ext_vector_type(8))) float v8f;

#define NN 100000
#define NE 1600000
#define DIN 32
#define DHID 16
#define DOUT 2

// ---------------- K0: zero workspace accumulators ----------------
__global__ __launch_bounds__(256) void k_zero(float* __restrict__ p, int n4) {
    int i = blockIdx.x * blockDim.x + threadIdx.x;
    if (i < n4) {
        *(float4*)(p + (long long)i * 4) = make_float4(0.f, 0.f, 0.f, 0.f);
    }
}

// ---------------- K1: scatter-add x -> agg1 (32 features, 8 thr/edge) ----------------
__global__ __launch_bounds__(256) void k_scatter32(const float* __restrict__ x,
                                                   const int* __restrict__ src,
                                                   const int* __restrict__ dst,
                                                   float* __restrict__ agg) {
    long long t = (long long)blockIdx.x * blockDim.x + threadIdx.x;
    int e = (int)(t >> 3);
    if (e >= NE) return;
    int c = ((int)t & 7) * 4;
    int s = src[e], d = dst[e];
    float4 v = *(const float4*)(x + (long long)s * DIN + c);
    float* a = agg + (long long)d * DIN + c;
    atomicAdd(a + 0, v.x);
    atomicAdd(a + 1, v.y);
    atomicAdd(a + 2, v.z);
    atomicAdd(a + 3, v.w);
}

// ---------------- K2: WMMA layer 1: h = relu(agg1@W1rel^T + x@W1root^T + b1) ----------------
// One wave per 16-node tile. C/D: lane l vgpr v -> (M = v + 8*(l>>4), N = l&15)
// A 16x4 f32: lane l vgpr v -> (M = l&15, K = 2*(l>>4) + v)
// B 4x16 f32: lane l vgpr v -> (K = 2*(l>>4) + v, N = l&15)
__global__ __launch_bounds__(256) void k_layer1_wmma(const float* __restrict__ x,
                                                     const float* __restrict__ agg1,
                                                     const float* __restrict__ w_rel,
                                                     const float* __restrict__ w_root,
                                                     const float* __restrict__ bias,
                                                     float* __restrict__ h) {
    const int lane = threadIdx.x & 31;
    const int wave = threadIdx.x >> 5;
    const int tile = blockIdx.x * (blockDim.x >> 5) + wave;
    const int NT = NN / 16;  // 6250
    if (tile >= NT) return;  // wave-uniform: EXEC stays all-ones for WMMA

    const int n  = lane & 15;   // column for B/C/D, row for A
    const int hi = lane >> 4;   // 0/1 half-wave
    const long long row0 = (long long)tile * 16;

    // accumulator preloaded with bias: C[m][n] = b[n] for all m
    float bn = bias[n];
    v8f c;
#pragma unroll
    for (int v = 0; v < 8; ++v) c[v] = bn;

    // agg1 @ W_rel^T  : B[k][n] = w_rel[n*DIN + k]
#pragma unroll
    for (int s = 0; s < 8; ++s) {
        int k = 4 * s + 2 * hi;
        v2f a = *(const v2f*)(agg1 + (row0 + n) * DIN + k);
        v2f b = *(const v2f*)(w_rel + n * DIN + k);
        c = __builtin_amdgcn_wmma_f32_16x16x4_f32(false, a, false, b, (short)0, c,
                                                  false, false);
    }
    // x @ W_root^T
#pragma unroll
    for (int s = 0; s < 8; ++s) {
        int k = 4 * s + 2 * hi;
        v2f a = *(const v2f*)(x + (row0 + n) * DIN + k);
        v2f b = *(const v2f*)(w_root + n * DIN + k);
        c = __builtin_amdgcn_wmma_f32_16x16x4_f32(false, a, false, b, (short)0, c,
                                                  false, false);
    }

    // fused ReLU + store: lane l vgpr v holds h[row0 + v + 8*hi][n]
#pragma unroll
    for (int v = 0; v < 8; ++v) {
        float val = c[v] > 0.f ? c[v] : 0.f;
        h[(row0 + v + 8 * hi) * DHID + n] = val;
    }
}

// ---------------- K3: scatter-add h -> agg2 (16 features, 4 thr/edge) ----------------
__global__ __launch_bounds__(256) void k_scatter16(const float* __restrict__ h,
                                                   const int* __restrict__ src,
                                                   const int* __restrict__ dst,
                                                   float* __restrict__ agg) {
    long long t = (long long)blockIdx.x * blockDim.x + threadIdx.x;
    int e = (int)(t >> 2);
    if (e >= NE) return;
    int c = ((int)t & 3) * 4;
    int s = src[e], d = dst[e];
    float4 v = *(const float4*)(h + (long long)s * DHID + c);
    float* a = agg + (long long)d * DHID + c;
    atomicAdd(a + 0, v.x);
    atomicAdd(a + 1, v.y);
    atomicAdd(a + 2, v.z);
    atomicAdd(a + 3, v.w);
}

// ---------------- K4: out = agg2@W2rel^T + h@W2root^T + b2  (D_OUT=2, scalar FMA) ----------------
__global__ __launch_bounds__(256) void k_layer2(const float* __restrict__ h,
                                                const float* __restrict__ agg2,
                                                const float* __restrict__ w_rel,
                                                const float* __restrict__ w_root,
                                                const float* __restrict__ bias,
                                                float* __restrict__ out) {
    int i = blockIdx.x * blockDim.x + threadIdx.x;
    if (i >= NN) return;
    const float* hv = h + (long long)i * DHID;
    const float* av = agg2 + (long long)i * DHID;
    float acc0 = bias[0];
    float acc1 = bias[1];
#pragma unroll
    for (int k = 0; k < DHID; ++k) {
        float hk = hv[k];
        float ak = av[k];
        acc0 = fmaf(ak, w_rel[k], acc0);
        acc0 = fmaf(hk, w_root[k], acc0);
        acc1 = fmaf(ak, w_rel[DHID + k], acc1);
        acc1 = fmaf(hk, w_root[DHID + k], acc1);
    }
    out[(long long)i * DOUT + 0] = acc0;
    out[(long long)i * DOUT + 1] = acc1;
}

extern "C" void kernel_launch(void* const* d_in, const int* in_sizes, int n_in,
                              void* d_out, int out_size, void* d_ws, size_t ws_size,
                              hipStream_t stream) {
    const float* x       = (const float*)d_in[0];
    const int*   ei      = (const int*)d_in[1];   // [2, NE]: row0 = src, row1 = dst
    const float* w1_rel  = (const float*)d_in[2];
    const float* w1_root = (const float*)d_in[3];
    const float* b1      = (const float*)d_in[4];
    const float* w2_rel  = (const float*)d_in[5];
    const float* w2_root = (const float*)d_in[6];
    const float* b2      = (const float*)d_in[7];
    float*       out     = (float*)d_out;

    const int* src = ei;
    const int* dst = ei + NE;

    // workspace layout: agg1 [NN*32] | h [NN*16] | agg2 [NN*16]
    float* agg1 = (float*)d_ws;
    float* h    = agg1 + (long long)NN * DIN;
    float* agg2 = h + (long long)NN * DHID;

    // K0: zero agg1 (contiguous) and agg2
    {
        int n4 = (NN * DIN) / 4;
        k_zero<<<(n4 + 255) / 256, 256, 0, stream>>>(agg1, n4);
        int n4b = (NN * DHID) / 4;
        k_zero<<<(n4b + 255) / 256, 256, 0, stream>>>(agg2, n4b);
    }

    // K1: edge scatter layer 1 (8 threads per edge)
    {
        long long threads = (long long)NE * 8;
        int blocks = (int)((threads + 255) / 256);
        k_scatter32<<<blocks, 256, 0, stream>>>(x, src, dst, agg1);
    }

    // K2: fused dense layer 1 via v_wmma_f32_16x16x4_f32 (1 wave / 16 nodes)
    {
        int tiles = NN / 16;                 // 6250
        int waves_per_block = 256 / 32;      // 8
        int blocks = (tiles + waves_per_block - 1) / waves_per_block;
        k_layer1_wmma<<<blocks, 256, 0, stream>>>(x, agg1, w1_rel, w1_root, b1, h);
    }

    // K3: edge scatter layer 2 (4 threads per edge)
    {
        long long threads = (long long)NE * 4;
        int blocks = (int)((threads + 255) / 256);
        k_scatter16<<<blocks, 256, 0, stream>>>(h, src, dst, agg2);
    }

    // K4: output layer (2 features -> scalar FMA, memory-bound)
    {
        int blocks = (NN + 255) / 256;
        k_layer2<<<blocks, 256, 0, stream>>>(h, agg2, w2_rel, w2_root, b2, out);
    }
}